// Laneline_loss_gflat_3D_47029891891852
// MI455X (gfx1250) — compile-verified
//
#include <hip/hip_runtime.h>
#include <hip/hip_bf16.h>
#include <math.h>

// Problem constants (match reference)
#define NN 256
#define XX 256
#define TT 3
#define KK 64
#define AA 193            // 3*K + 1
#define EPSF 1e-9f

#define IDX(n, x, t, a) ((((n) * XX + (x)) * TT + (t)) * AA + (a))

typedef __attribute__((ext_vector_type(2))) float v2f;
typedef __attribute__((ext_vector_type(8))) float v8f;

// ---------------------------------------------------------------------------
// Wave32 reduction on the matrix pipe: V_WMMA_F32_16X16X4_F32 with B = ones.
// A 16x4 f32 tile layout: lanes 0-15 hold rows M=0..15 (VGPR0=K0, VGPR1=K1),
// lanes 16-31 hold K2/K3.  Putting lane-partial p in a.x and 0 in a.y gives
// D[m,n] = p_m + p_{m+16}.  Each lane then sums its 8 C VGPRs:
//   lanes 0-15  all hold  halfA = sum_{m=0..7}  (p_m + p_{m+16})
//   lanes 16-31 all hold  halfB = sum_{m=8..15} (p_m + p_{m+16})
// halfA + halfB = full 32-lane sum.  We skip any shuffle: lane 0 and lane 16
// write their halves to LDS and thread 0 combines — no ds_bpermute needed.
// EXEC must be all-ones at the call site (no divergence there).
// ---------------------------------------------------------------------------
__device__ __forceinline__ float wave_half_sum(float p) {
  v2f a; a.x = p;    a.y = 0.0f;
  v2f b; b.x = 1.0f; b.y = 1.0f;
  v8f c = {};
  c = __builtin_amdgcn_wmma_f32_16x16x4_f32(
      /*neg_a=*/false, a, /*neg_b=*/false, b,
      /*c_mod=*/(short)0, c, /*reuse_a=*/false, /*reuse_b=*/false);
  return c[0] + c[1] + c[2] + c[3] + c[4] + c[5] + c[6] + c[7];
}

// Block reduction: each wave contributes two half-sums (lane 0, lane 16).
// Caller must pass a __shared__ float[16] buffer (supports up to 8 waves).
__device__ __forceinline__ void block_accumulate(float p, float* sred,
                                                 float scale,
                                                 float* __restrict__ out) {
  float h = wave_half_sum(p);
  int lane = threadIdx.x & 31;
  int wid  = threadIdx.x >> 5;
  if (lane == 0)       sred[wid]     = h;   // halfA
  else if (lane == 16) sred[8 + wid] = h;   // halfB
  __syncthreads();
  if (threadIdx.x == 0) {
    float s = 0.0f;
    int nw = blockDim.x >> 5;
    for (int w = 0; w < nw; ++w) s += sred[w] + sred[8 + w];
    atomicAdd(out, scale * s);
  }
}

// ---------------------------------------------------------------------------
// Kernel 0: zero the output accumulator (d_out is poisoned by the harness).
// ---------------------------------------------------------------------------
__global__ void zero_out_kernel(float* __restrict__ out, int nelem) {
  int i = blockIdx.x * blockDim.x + threadIdx.x;
  if (i < nelem) out[i] = 0.0f;
}

// ---------------------------------------------------------------------------
// Kernel 1: elementwise losses (loss0 + loss1 + loss2), bandwidth bound.
// One thread per (group, k); k = i & 63 => coalesced b32 loads per wave.
// ---------------------------------------------------------------------------
__global__ void __launch_bounds__(256)
loss_elem_kernel(const float* __restrict__ pred,
                 const float* __restrict__ gt,
                 float* __restrict__ out,
                 int total) {
  float acc = 0.0f;
  const float invK = 1.0f / (float)KK;
  for (int i = blockIdx.x * blockDim.x + threadIdx.x; i < total;
       i += gridDim.x * blockDim.x) {
    int g = i >> 6;          // (n*X + x)*T + t
    int k = i & 63;
    int base = g * AA;

    // loss0: vis BCE, divided by K
    float pv = pred[base + 2 * KK + k];
    float gv = gt[base + 2 * KK + k];
    float t0 = -(gv * __logf(pv + EPSF) +
                 (1.0f - gv + EPSF) * __logf(1.0f - pv + EPSF)) * invK;

    // loss2: |gt_class * gt_vis * (pred_Xoff - gt_Xoff)|
    float px = pred[base + k];
    float gx = gt[base + k];
    float gc = gt[base + 3 * KK];     // broadcast within group (L0 hit)
    float t2 = fabsf(gc * gv * (px - gx));

    acc += t0 + t2;

    // loss1: class BCE, one term per group
    if (k == 0) {
      float pc = pred[base + 3 * KK];
      acc += -(gc * __logf(pc + EPSF) +
               (1.0f - gc) * __logf(1.0f - pc + EPSF));
    }
  }

  __shared__ float sred[16];
  block_accumulate(acc, sred, 1.0f, out);
}

// ---------------------------------------------------------------------------
// Kernel 2: per-batch geometric losses (lw, lh).  One block per batch n.
// Replicates the reference's compaction: selected anchors first (stable
// ascending), kept columns (col_sum >= n_rows or k < 5) first (stable
// ascending); only the compacted-valid region contributes, so we iterate the
// kept lists directly.  Output only needs 5*(lw+lh), so a single scalar is
// reduced through the matrix pipe.
// ---------------------------------------------------------------------------
__global__ void __launch_bounds__(256)
loss_geom_kernel(const float* __restrict__ pred,
                 const float* __restrict__ gt,
                 const float* __restrict__ hcam,
                 const float* __restrict__ ax_g,
                 const float* __restrict__ ay_g,
                 const float* __restrict__ xstd_g,
                 const float* __restrict__ zstd_g,
                 float* __restrict__ out) {
  const int n = blockIdx.x;
  const int tid = threadIdx.x;

  __shared__ float s_ax[XX];
  __shared__ float s_ay[KK], s_xstd[KK], s_zstd[KK];
  __shared__ int   s_sel[XX];
  __shared__ int   s_cols[KK];
  __shared__ int   s_nsel, s_m;
  __shared__ unsigned char s_flag[XX];
  __shared__ unsigned char s_keep[KK];
  __shared__ float s_part[4][KK];
  __shared__ float s_red[16];

  // constants to LDS
  s_ax[tid] = ax_g[tid];
  if (tid < KK) {
    s_ay[tid]   = ay_g[tid];
    s_xstd[tid] = xstd_g[tid];
    s_zstd[tid] = zstd_g[tid];
  }

  // sel = gt_class[:, :, 0, 0] > 0
  s_flag[tid] = (gt[IDX(n, tid, 0, 3 * KK)] > 0.0f) ? 1 : 0;
  __syncthreads();

  if (tid == 0) {
    int c = 0;
    for (int x = 0; x < XX; ++x)
      if (s_flag[x]) s_sel[c++] = x;
    s_nsel = c;
  }
  __syncthreads();

  const int nsel   = s_nsel;
  const int n_rows = nsel * TT;

  // column sums of gt_vis over valid (selected) rows
  {
    int k = tid & 63, grp = tid >> 6;
    float cs = 0.0f;
    for (int i = grp; i < nsel; i += 4) {
      int a  = s_sel[i];
      int b0 = IDX(n, a, 0, 2 * KK + k);
      cs += gt[b0] + gt[b0 + AA] + gt[b0 + 2 * AA];
    }
    s_part[grp][k] = cs;
  }
  __syncthreads();
  if (tid < KK) {
    float c = s_part[0][tid] + s_part[1][tid] + s_part[2][tid] + s_part[3][tid];
    s_keep[tid] = ((c >= (float)n_rows) || (tid < 5)) ? 1 : 0;
  }
  __syncthreads();
  if (tid == 0) {
    int m = 0;
    for (int k = 0; k < KK; ++k)
      if (s_keep[k]) s_cols[m++] = k;
    s_m = m;
  }
  __syncthreads();

  const int   m     = s_m;          // >= 5 whenever it matters (k<5 forced)
  const float inv_h = 1.0f / hcam[n];
  float acc = 0.0f;                 // accumulates lw + lh jointly

  const int nd = n_rows - 1;        // valid diff-row pairs: r, r+1 in [0, n_rows)
  for (int r = tid; r < nd; r += blockDim.x) {
    const int a0 = s_sel[r / TT],   t0 = r % TT;
    const int r1 = r + 1;
    const int a1 = s_sel[r1 / TT],  t1 = r1 % TT;
    const int base0 = IDX(n, a0, t0, 0);
    const int base1 = IDX(n, a1, t1, 0);
    const float axa0 = s_ax[a0], axa1 = s_ax[a1];

    // width0: |gt_Xg(r+1, k=0) - gt_Xg(r, k=0)|
    float g0 = gt[base0 + 0] * s_xstd[0] + axa0;
    float g1 = gt[base1 + 0] * s_xstd[0] + axa1;
    float wprev = fabsf(g1 - g0);     // ori[r, 0]

    float prevLane = 0.0f, prevY = 0.0f;
    for (int j = 0; j < m; ++j) {
      const int c = s_cols[j];
      // row r quantities
      float zsA   = pred[base0 + KK + c] * s_zstd[c];
      float scA   = 1.0f - zsA * inv_h;
      float laneA = scA * (gt[base0 + c] * s_xstd[c] + axa0);
      float YA    = scA * s_ay[c];
      // row r+1 quantities (only lane and Z needed)
      float zsB   = pred[base1 + KK + c] * s_zstd[c];
      float scB   = 1.0f - zsB * inv_h;
      float laneB = scB * (gt[base1 + c] * s_xstd[c] + axa1);

      float lY, l;
      if (j == 0) {
        // lY_0 = l_0 = |Y(r, c1) - Y(r, c0)|  (peek next kept column; m >= 2)
        int   c1  = s_cols[1];
        float zs1 = pred[base0 + KK + c1] * s_zstd[c1];
        float sc1 = 1.0f - zs1 * inv_h;
        float Y1  = sc1 * s_ay[c1];
        lY = fabsf(Y1 - YA);
        l  = lY;
      } else {
        float dl = laneA - prevLane;
        float dy = YA - prevY;
        lY = fabsf(dy);
        l  = sqrtf(dl * dl + dy * dy);
      }

      float w = fabsf(laneB - laneA) * lY / (l + EPSF);
      acc += fabsf(w - wprev);         // werr[r, j]  (lw part)
      wprev = w;
      acc += fabsf(zsB - zsA);         // dZ[r, j]    (lh part)

      prevLane = laneA;
      prevY    = YA;
    }
  }

  // block reduction through the matrix pipe; contribution is 5*(lw+lh)
  block_accumulate(acc, s_red, 5.0f, out);
}

// ---------------------------------------------------------------------------
// Launch
// ---------------------------------------------------------------------------
extern "C" void kernel_launch(void* const* d_in, const int* in_sizes, int n_in,
                              void* d_out, int out_size, void* d_ws, size_t ws_size,
                              hipStream_t stream) {
  const float* pred = (const float*)d_in[0];
  const float* gt   = (const float*)d_in[1];
  const float* hcam = (const float*)d_in[2];
  const float* ax   = (const float*)d_in[3];
  const float* ay   = (const float*)d_in[4];
  const float* xstd = (const float*)d_in[5];
  const float* zstd = (const float*)d_in[6];
  float* out = (float*)d_out;

  (void)in_sizes; (void)n_in; (void)d_ws; (void)ws_size;

  zero_out_kernel<<<1, 32, 0, stream>>>(out, out_size);

  const int total = NN * XX * TT * KK;   // 12.58M element slots
  loss_elem_kernel<<<4096, 256, 0, stream>>>(pred, gt, out, total);

  loss_geom_kernel<<<NN, 256, 0, stream>>>(pred, gt, hcam, ax, ay, xstd, zstd, out);
}